// GraphSAGENet_13099650253557
// MI455X (gfx1250) — compile-verified
//
#include <hip/hip_runtime.h>
#include <math.h>

// GraphSAGE (6x SAGEConv(max)+GraphNorm+LeakyReLU, then SAGEConv->3, tanh*0.5)
// MI455X / gfx1250, wave32. fp32 throughout; matrix ops via V_WMMA_F32_16X16X4_F32.
// Aggregation via one-time CSR build + wave-per-node register max-reduce
// (avoids ~1.4G float atomics that would dominate runtime).

#define NFC 128
#define NEG_SLOPE 0.02f
#define EPS_GN 1e-5f

typedef __attribute__((ext_vector_type(2))) float v2f;
typedef __attribute__((ext_vector_type(8))) float v8f;

// ============================ CSR build (once) =============================

__global__ void zero_int(int* __restrict__ p, int n) {
  int i = blockIdx.x * blockDim.x + threadIdx.x;
  if (i < n) p[i] = 0;
}

__global__ void count_edges(const int* __restrict__ ei, int* __restrict__ cnt,
                            int nE) {
  int e = blockIdx.x * blockDim.x + threadIdx.x;
  if (e < nE) atomicAdd(&cnt[ei[nE + e]], 1);
}

// Pass 1: per-block (1024 elems) exclusive scan of cnt -> rowptr; block totals.
__global__ void scan_pass1(const int* __restrict__ cnt, int* __restrict__ rowptr,
                           int* __restrict__ blockSums, int n) {
  __shared__ int sm[256];
  int tid = threadIdx.x;
  int i0 = blockIdx.x * 1024 + tid * 4;
  int c0 = (i0 + 0 < n) ? cnt[i0 + 0] : 0;
  int c1 = (i0 + 1 < n) ? cnt[i0 + 1] : 0;
  int c2 = (i0 + 2 < n) ? cnt[i0 + 2] : 0;
  int c3 = (i0 + 3 < n) ? cnt[i0 + 3] : 0;
  int tsum = c0 + c1 + c2 + c3;
  sm[tid] = tsum;
  __syncthreads();
  for (int off = 1; off < 256; off <<= 1) {
    int t = (tid >= off) ? sm[tid - off] : 0;
    __syncthreads();
    sm[tid] += t;
    __syncthreads();
  }
  int excl = sm[tid] - tsum;
  if (i0 + 0 < n) rowptr[i0 + 0] = excl;
  if (i0 + 1 < n) rowptr[i0 + 1] = excl + c0;
  if (i0 + 2 < n) rowptr[i0 + 2] = excl + c0 + c1;
  if (i0 + 3 < n) rowptr[i0 + 3] = excl + c0 + c1 + c2;
  if (tid == 255) blockSums[blockIdx.x] = excl + tsum;  // block total
}

// Pass 2: exclusive scan of block totals (nb <= 256; 100k nodes -> 98 blocks).
__global__ void scan_pass2(int* __restrict__ blockSums, int nb) {
  __shared__ int sm[256];
  int tid = threadIdx.x;
  int v = (tid < nb) ? blockSums[tid] : 0;
  sm[tid] = v;
  __syncthreads();
  for (int off = 1; off < 256; off <<= 1) {
    int t = (tid >= off) ? sm[tid - off] : 0;
    __syncthreads();
    sm[tid] += t;
    __syncthreads();
  }
  if (tid < nb) blockSums[tid] = sm[tid] - v;
}

// Pass 3: add block offsets; rowptr[n] = nE.
__global__ void scan_pass3(int* __restrict__ rowptr,
                           const int* __restrict__ blockSums, int n, int nE) {
  int i = blockIdx.x * blockDim.x + threadIdx.x;
  if (i < n) rowptr[i] += blockSums[i >> 10];
  if (i == 0) rowptr[n] = nE;
}

__global__ void copy_int(const int* __restrict__ src, int* __restrict__ dst,
                         int n) {
  int i = blockIdx.x * blockDim.x + threadIdx.x;
  if (i < n) dst[i] = src[i];
}

// Bucket-fill: srcs sorted by dst. Order within a segment is racy but the
// downstream max-reduce is order-insensitive -> deterministic output.
__global__ void fill_csr(const int* __restrict__ ei, int* __restrict__ fillpos,
                         int* __restrict__ srcs, int nE) {
  int e = blockIdx.x * blockDim.x + threadIdx.x;
  if (e < nE) {
    int p = atomicAdd(&fillpos[ei[nE + e]], 1);
    srcs[p] = ei[e];
  }
}

// ===================== segment max: wave per destination ====================
// Lanes cover 128 channels as float4 (512B contiguous gather per src row).
// srcs loaded coalesced 32-at-a-time, broadcast with __shfl. Max in registers,
// one streaming store. Empty segments write 0 (matches where(isfinite,m,0)).
__global__ void aggregate_max(const float* __restrict__ h,
                              const int* __restrict__ rowptr,
                              const int* __restrict__ srcs,
                              float* __restrict__ m, int nN) {
  int gid = blockIdx.x * blockDim.x + threadIdx.x;
  int node = gid >> 5;
  int lane = gid & 31;
  if (node >= nN) return;
  int beg = rowptr[node];
  int end = rowptr[node + 1];
  float4 acc = make_float4(-INFINITY, -INFINITY, -INFINITY, -INFINITY);
  int idx = beg;
  while (idx < end) {
    int nload = end - idx;
    nload = (nload > 32) ? 32 : nload;
    int sl = (idx + lane < end) ? srcs[idx + lane] : 0;
    for (int j = 0; j < nload; ++j) {
      int s = __shfl(sl, j, 32);
      float4 v = ((const float4*)(h + (size_t)s * NFC))[lane];
      acc.x = fmaxf(acc.x, v.x);
      acc.y = fmaxf(acc.y, v.y);
      acc.z = fmaxf(acc.z, v.z);
      acc.w = fmaxf(acc.w, v.w);
    }
    idx += nload;
  }
  if (beg == end) acc = make_float4(0.0f, 0.0f, 0.0f, 0.0f);
  ((float4*)(m + (size_t)node * NFC))[lane] = acc;
}

// ---------------------------------------------- SAGE linear: m@Wl^T + h@Wr^T + b
// One wave per 16-node tile; computes all 128 output channels (8 col tiles)
// so m/h rows are read exactly once. fp32 WMMA, K accumulated 4 at a time.
// A 16x4 f32: row = lane&15, K = 2*(lane>>4)+{0,1} -> float2 load.
// B 4x16  f32: col = lane&15, same K pattern -> float2 of Wl[c*128+k]
//   (B[k][c] = Wl[c][k], i.e. out = m @ Wl^T).
// C/D: col = lane&15, row = i + 8*(lane>>4) for acc element i.
__global__ void sage_gemm_wmma(const float* __restrict__ h,
                               const float* __restrict__ m,
                               const float* __restrict__ Wl,
                               const float* __restrict__ Wr,
                               const float* __restrict__ bias,
                               float* __restrict__ out, int ntiles) {
  int wave = (int)((blockIdx.x * blockDim.x + threadIdx.x) >> 5);
  if (wave >= ntiles) return;                 // wave-uniform: EXEC stays all-1s
  int lane = threadIdx.x & 31;
  int r = lane & 15;
  int kh = (lane >> 4) * 2;
  int node = wave * 16 + r;
  const float* hrow = h + (size_t)node * NFC;
  const float* mrow = m + (size_t)node * NFC;

  v8f acc[8];
#pragma unroll
  for (int t = 0; t < 8; ++t) acc[t] = (v8f)0.0f;

  for (int k0 = 0; k0 < NFC; k0 += 4) {
    int kk = k0 + kh;
    v2f am = *(const v2f*)(mrow + kk);   // CSR aggregate is always finite
    v2f ah = *(const v2f*)(hrow + kk);
#pragma unroll
    for (int t = 0; t < 8; ++t) {
      int c = t * 16 + r;
      v2f bW = *(const v2f*)(Wl + (size_t)c * NFC + kk);
      v2f bR = *(const v2f*)(Wr + (size_t)c * NFC + kk);
      acc[t] = __builtin_amdgcn_wmma_f32_16x16x4_f32(
          false, am, false, bW, (short)0, acc[t], false, false);
      acc[t] = __builtin_amdgcn_wmma_f32_16x16x4_f32(
          false, ah, false, bR, (short)0, acc[t], false, false);
    }
  }

  int rbase = wave * 16 + (lane >> 4) * 8;
#pragma unroll
  for (int t = 0; t < 8; ++t) {
    int c = t * 16 + r;
    float b = bias[c];
#pragma unroll
    for (int i = 0; i < 8; ++i) {
      out[(size_t)(rbase + i) * NFC + c] = acc[t][i] + b;
    }
  }
}

// ------------------------------------------------- GraphNorm stats (one pass)
__global__ void zero_stats(float* __restrict__ stats) {
  int t = threadIdx.x;
  if (t < 256) stats[t] = 0.0f;  // [0..127]=sum, [128..255]=sumsq
}

__global__ void reduce_stats(const float* __restrict__ h,
                             float* __restrict__ stats, int total) {
  __shared__ float ls[256];
  __shared__ float ls2[256];
  int tid = threadIdx.x;
  int gid = blockIdx.x * 256 + tid;
  int stride = gridDim.x * 256;      // multiple of 128 -> channel is invariant
  float s = 0.0f, s2 = 0.0f;
  for (int i = gid; i < total; i += stride) {
    float v = h[i];
    s += v;
    s2 += v * v;
  }
  ls[tid] = s;
  ls2[tid] = s2;
  __syncthreads();
  if (tid < 128) {
    s = ls[tid] + ls[tid + 128];
    s2 = ls2[tid] + ls2[tid + 128];
    atomicAdd(&stats[tid], s);          // channel == tid (stride multiple of 128)
    atomicAdd(&stats[128 + tid], s2);
  }
}

// var = E[h^2] + mean^2*(a^2 - 2a);  out = h*scale + shift
__global__ void compute_scale_shift(float* __restrict__ stats,
                                    const float* __restrict__ gamma,
                                    const float* __restrict__ beta,
                                    const float* __restrict__ alpha, int nN) {
  int c = threadIdx.x;
  if (c >= NFC) return;
  float invN = 1.0f / (float)nN;
  float mean = stats[c] * invN;
  float ex2 = stats[128 + c] * invN;
  float a = alpha[c];
  float var = ex2 + mean * mean * (a * a - 2.0f * a);
  float rs = rsqrtf(var + EPS_GN);
  float sc = gamma[c] * rs;
  stats[256 + c] = sc;
  stats[384 + c] = beta[c] - sc * a * mean;
}

__global__ void apply_norm_leaky(float* __restrict__ h,
                                 const float* __restrict__ stats, int n4) {
  int i = blockIdx.x * blockDim.x + threadIdx.x;
  if (i >= n4) return;
  int c = (i * 4) & (NFC - 1);             // 4 consecutive channels, same row
  float4 v = ((float4*)h)[i];
  float4 sc = *(const float4*)(stats + 256 + c);
  float4 sh = *(const float4*)(stats + 384 + c);
  v.x = v.x * sc.x + sh.x; v.x = (v.x > 0.0f) ? v.x : NEG_SLOPE * v.x;
  v.y = v.y * sc.y + sh.y; v.y = (v.y > 0.0f) ? v.y : NEG_SLOPE * v.y;
  v.z = v.z * sc.z + sh.z; v.z = (v.z > 0.0f) ? v.z : NEG_SLOPE * v.z;
  v.w = v.w * sc.w + sh.w; v.w = (v.w > 0.0f) ? v.w : NEG_SLOPE * v.w;
  ((float4*)h)[i] = v;
}

// ------------------------------------------------- final layer: 128 -> 3, tanh
__global__ void final_sage(const float* __restrict__ h,
                           const float* __restrict__ m,
                           const float* __restrict__ WlO,
                           const float* __restrict__ blO,
                           const float* __restrict__ WrO,
                           float* __restrict__ out, int nN) {
  int gid = blockIdx.x * blockDim.x + threadIdx.x;
  int node = gid >> 5;
  int lane = gid & 31;
  if (node >= nN) return;
  const float* hr = h + (size_t)node * NFC;
  const float* mr = m + (size_t)node * NFC;
  float a0 = 0.0f, a1 = 0.0f, a2 = 0.0f;
  for (int k = lane; k < NFC; k += 32) {
    float mv = mr[k];
    float hv = hr[k];
    a0 += mv * WlO[k]       + hv * WrO[k];
    a1 += mv * WlO[128 + k] + hv * WrO[128 + k];
    a2 += mv * WlO[256 + k] + hv * WrO[256 + k];
  }
#pragma unroll
  for (int off = 16; off > 0; off >>= 1) {
    a0 += __shfl_xor(a0, off, 32);
    a1 += __shfl_xor(a1, off, 32);
    a2 += __shfl_xor(a2, off, 32);
  }
  if (lane == 0) {
    out[node * 3 + 0] = tanhf(a0 + blO[0]) * 0.5f;
    out[node * 3 + 1] = tanhf(a1 + blO[1]) * 0.5f;
    out[node * 3 + 2] = tanhf(a2 + blO[2]) * 0.5f;
  }
}

// ---------------------------------------------------------------------------
extern "C" void kernel_launch(void* const* d_in, const int* in_sizes, int n_in,
                              void* d_out, int out_size, void* d_ws, size_t ws_size,
                              hipStream_t stream) {
  const float* x     = (const float*)d_in[0];
  const int*   ei    = (const int*)d_in[1];
  const float* Wl    = (const float*)d_in[2];
  const float* bl    = (const float*)d_in[3];
  const float* Wr    = (const float*)d_in[4];
  const float* WlO   = (const float*)d_in[5];
  const float* blO   = (const float*)d_in[6];
  const float* WrO   = (const float*)d_in[7];
  const float* gamma = (const float*)d_in[8];
  const float* beta  = (const float*)d_in[9];
  const float* alpha = (const float*)d_in[10];

  int nN = in_sizes[0] / NFC;      // 100000
  int nE = in_sizes[1] / 2;        // 1600000

  float* ws    = (float*)d_ws;
  size_t big   = (size_t)nN * NFC;
  float* hA    = ws;
  float* hB    = ws + big;
  float* mbuf  = ws + 2 * big;
  float* stats = ws + 3 * big;            // 512 floats: sum,sumsq,scale,shift
  int* rowptr  = (int*)(stats + 512);     // nN+1
  int* fillpos = rowptr + (nN + 1);       // nN (doubles as counts)
  int* srcs    = fillpos + nN;            // nE
  int* bsum    = srcs + nE;               // scan block totals

  int total = (int)big;
  int n4 = total / 4;
  dim3 blk(256);
  int nodeB  = (nN + 255) / 256;
  int edgeB  = (nE + 255) / 256;
  int scanB  = (nN + 1023) / 1024;           // <= 256 for N <= 262144
  int ntiles = nN / 16;                      // N divisible by 16 (100000)
  int gemmB  = (ntiles * 32 + 255) / 256;    // one wave per 16-node tile
  int waveB  = (nN * 32 + 255) / 256;        // one wave per node
  int fillB  = (n4 + 255) / 256;

  // ---- build CSR (dst-sorted incoming edges), once: edges reused 7x ----
  zero_int<<<nodeB, blk, 0, stream>>>(fillpos, nN);
  count_edges<<<edgeB, blk, 0, stream>>>(ei, fillpos, nE);
  scan_pass1<<<scanB, blk, 0, stream>>>(fillpos, rowptr, bsum, nN);
  scan_pass2<<<1, blk, 0, stream>>>(bsum, scanB);
  scan_pass3<<<nodeB, blk, 0, stream>>>(rowptr, bsum, nN, nE);
  copy_int<<<nodeB, blk, 0, stream>>>(rowptr, fillpos, nN);
  fill_csr<<<edgeB, blk, 0, stream>>>(ei, fillpos, srcs, nE);

  // ---- 6 hidden layers ----
  const float* cur = x;
  for (int layer = 0; layer < 6; ++layer) {
    float* hout = (layer & 1) ? hB : hA;
    aggregate_max<<<waveB, blk, 0, stream>>>(cur, rowptr, srcs, mbuf, nN);
    sage_gemm_wmma<<<gemmB, blk, 0, stream>>>(
        cur, mbuf, Wl + (size_t)layer * NFC * NFC, Wr + (size_t)layer * NFC * NFC,
        bl + (size_t)layer * NFC, hout, ntiles);
    zero_stats<<<1, 256, 0, stream>>>(stats);
    reduce_stats<<<1024, blk, 0, stream>>>(hout, stats, total);
    compute_scale_shift<<<1, 128, 0, stream>>>(
        stats, gamma + (size_t)layer * NFC, beta + (size_t)layer * NFC,
        alpha + (size_t)layer * NFC, nN);
    apply_norm_leaky<<<fillB, blk, 0, stream>>>(hout, stats, n4);
    cur = hout;
  }

  // ---- output layer: max-aggregate then 128->3 + tanh*0.5 ----
  aggregate_max<<<waveB, blk, 0, stream>>>(cur, rowptr, srcs, mbuf, nN);
  final_sage<<<waveB, blk, 0, stream>>>(cur, mbuf, WlO, blO, WrO,
                                        (float*)d_out, nN);
}